// QRStep_69114613730268
// MI455X (gfx1250) — compile-verified
//
#include <hip/hip_runtime.h>
#include <math.h>

// ---------------------------------------------------------------------------
// QRStep for MI455X (gfx1250, wave32, WMMA).
//   curr = init_map^T @ op @ init_map          (2 WMMA-f32 GEMMs)
//   Q    = blocked classical Gram-Schmidt      (per-panel WMMA GEMMs + 1-WG GS)
//   out  = init_map @ Q                        (1 WMMA-f32 GEMM)
// FP32 throughout (V_WMMA_F32_16X16X4_F32) to match reference numerics.
// B tiles staged via GLOBAL_LOAD_ASYNC_TO_LDS_B128 when in-bounds (ASYNCcnt).
// Workspace layout (floats): curr[n*n] | Q[n*n] | S[n*PB] | R[n*PB]  (~34 MB)
// ---------------------------------------------------------------------------

typedef float v2f __attribute__((ext_vector_type(2)));
typedef float v8f __attribute__((ext_vector_type(8)));
typedef int   v4i __attribute__((ext_vector_type(4)));

#define BM 64
#define BN 128
#define KB 32
#define BPAD 4   // B-tile pad (keeps 16B chunks aligned & contiguous)
#define PB 16    // Gram-Schmidt panel width (multiple of WMMA N=16)

#if defined(__gfx1250__) && __has_builtin(__builtin_amdgcn_global_load_async_to_lds_b128)
#define ASYNC_LDS 1
typedef __attribute__((address_space(1))) v4i GV4;   // global int4
typedef __attribute__((address_space(3))) v4i LV4;   // LDS int4
#endif

__device__ __forceinline__ void wait_async_zero() {
#if defined(ASYNC_LDS)
#if __has_builtin(__builtin_amdgcn_s_wait_asynccnt)
    __builtin_amdgcn_s_wait_asynccnt(0);
#else
    asm volatile("s_wait_asynccnt 0x0" ::: "memory");
#endif
#endif
}

// D[M x N] = (mode==0)  opA @ B
//          = (mode==1)  addC - opA @ B
// opA[m][k] = transA ? A[k*lda+m] : A[m*lda+k];  all row-major.
__global__ __launch_bounds__(256)
void gemm_wmma_f32(float* __restrict__ D, int ldd,
                   const float* __restrict__ A, int lda, int transA,
                   const float* __restrict__ B, int ldb,
                   const float* __restrict__ addC, int ldc,
                   int M, int N, int K, int mode)
{
    __shared__ float As[BM][KB + 1];
    __shared__ float Bs[KB][BN + BPAD];

    const int tid  = threadIdx.x;
    const int lane = tid & 31;
    const int wave = tid >> 5;     // 8 waves: 2 (M) x 4 (N)
    const int wm   = wave >> 2;    // 0..1  -> 32 rows each
    const int wn   = wave & 3;     // 0..3  -> 32 cols each
    const int hh   = lane >> 4;    // lane half: K pair (A/B), row pair (C)
    const int lr   = lane & 15;

    const int m0 = blockIdx.y * BM;
    const int n0 = blockIdx.x * BN;

    v8f acc00 = {}, acc01 = {}, acc10 = {}, acc11 = {};

    const int ktiles = (K + KB - 1) / KB;
    for (int kt = 0; kt < ktiles; ++kt) {
        const int k0 = kt * KB;

        // ---- stage A tile 64x32 into LDS (8 elems/thread), zero-fill OOB
        #pragma unroll
        for (int t = 0; t < 8; ++t) {
            int l = tid + t * 256;
            int r = l >> 5, c = l & 31;
            int gm = m0 + r, gk = k0 + c;
            float v = 0.0f;
            if (gm < M && gk < K)
                v = transA ? A[(size_t)gk * lda + gm] : A[(size_t)gm * lda + gk];
            As[r][c] = v;
        }

        // ---- stage B tile 32x128 into LDS
        const bool fullB = (k0 + KB <= K) && (n0 + BN <= N);
#if defined(ASYNC_LDS)
        if (fullB) {
            // 1024 16B chunks: 4 async b128 transfers per thread, no VGPR data
            #pragma unroll
            for (int t = 0; t < 4; ++t) {
                int idx = tid + t * 256;
                int r   = idx >> 5;            // 0..31 (K row)
                int c4  = (idx & 31) << 2;     // 0,4,...,124 (N col, 16B chunk)
                const float* gp = B + (size_t)(k0 + r) * ldb + (n0 + c4);
                __builtin_amdgcn_global_load_async_to_lds_b128(
                    (GV4*)gp, (LV4*)&Bs[r][c4], 0, 0);
            }
            wait_async_zero();
        } else
#endif
        {
            #pragma unroll
            for (int t = 0; t < 16; ++t) {
                int l = tid + t * 256;
                int r = l >> 7, c = l & 127;
                int gk = k0 + r, gn = n0 + c;
                float v = 0.0f;
                if (gk < K && gn < N)
                    v = B[(size_t)gk * ldb + gn];
                Bs[r][c] = v;
            }
        }

        // prefetch next A K-tile (global_prefetch_b8; cacheline granular)
        if (kt + 1 < ktiles) {
            int r = tid >> 5, c = tid & 31;
            int gm = m0 + r, gk = k0 + KB + c;
            if (gm < M && gk < K)
                __builtin_prefetch(transA ? &A[(size_t)gk * lda + gm]
                                          : &A[(size_t)gm * lda + gk], 0, 0);
        }
        __syncthreads();

        // ---- 8 K-steps x 4 WMMAs: each wave owns a 32x32 output
        #pragma unroll
        for (int kk = 0; kk < KB; kk += 4) {
            v2f a0, a1, b0, b1;
            const int am = wm * 32 + lr;
            a0.x = As[am][kk + hh * 2 + 0];
            a0.y = As[am][kk + hh * 2 + 1];
            a1.x = As[am + 16][kk + hh * 2 + 0];
            a1.y = As[am + 16][kk + hh * 2 + 1];
            const int bn = wn * 32 + lr;
            b0.x = Bs[kk + hh * 2 + 0][bn];
            b0.y = Bs[kk + hh * 2 + 1][bn];
            b1.x = Bs[kk + hh * 2 + 0][bn + 16];
            b1.y = Bs[kk + hh * 2 + 1][bn + 16];
            acc00 = __builtin_amdgcn_wmma_f32_16x16x4_f32(false, a0, false, b0,
                                                          (short)0, acc00, false, false);
            acc01 = __builtin_amdgcn_wmma_f32_16x16x4_f32(false, a0, false, b1,
                                                          (short)0, acc01, false, false);
            acc10 = __builtin_amdgcn_wmma_f32_16x16x4_f32(false, a1, false, b0,
                                                          (short)0, acc10, false, false);
            acc11 = __builtin_amdgcn_wmma_f32_16x16x4_f32(false, a1, false, b1,
                                                          (short)0, acc11, false, false);
        }
        __syncthreads();
    }

    // ---- epilogue: C/D layout -> VGPR r: lanes 0-15 M=r, lanes 16-31 M=8+r
    auto store_tile = [&](const v8f& acc, int ti, int tj) {
        #pragma unroll
        for (int r = 0; r < 8; ++r) {
            int m  = m0 + wm * 32 + ti * 16 + hh * 8 + r;
            int nn = n0 + wn * 32 + tj * 16 + lr;
            if (m < M && nn < N) {
                float v = acc[r];
                if (mode == 1) v = addC[(size_t)m * ldc + nn] - v;
                D[(size_t)m * ldd + nn] = v;
            }
        }
    };
    store_tile(acc00, 0, 0);
    store_tile(acc01, 0, 1);
    store_tile(acc10, 1, 0);
    store_tile(acc11, 1, 1);
}

// Intra-panel classical Gram-Schmidt: one workgroup, sequential over PB cols.
// R holds panel residual vs all previous panels (original-column projections),
// dots against within-panel Q columns use the ORIGINAL curr columns (CGS).
__global__ __launch_bounds__(256)
void panel_gs(const float* __restrict__ R, const float* __restrict__ curr,
              float* __restrict__ Q, int col0, int pb, int n)
{
    __shared__ float red[256];
    __shared__ float dots[32];
    const int tid = threadIdx.x;
    const int per = n / 256;    // 8 for n=2048
    float v[8];

    for (int c = 0; c < pb; ++c) {
        // dots[j] = Q[:,col0+j] . curr[:,col0+c], j < c (within panel)
        for (int j = 0; j < c; ++j) {
            float p = 0.0f;
            for (int t = 0; t < per; ++t) {
                int i = tid + t * 256;
                p += Q[(size_t)i * n + col0 + j] * curr[(size_t)i * n + col0 + c];
            }
            red[tid] = p; __syncthreads();
            for (int s = 128; s > 0; s >>= 1) {
                if (tid < s) red[tid] += red[tid + s];
                __syncthreads();
            }
            if (tid == 0) dots[j] = red[0];
            __syncthreads();
        }
        // v = R[:,c] - sum_j q_j * dots[j];  accumulate ||v||^2
        float nrm = 0.0f;
        for (int t = 0; t < per; ++t) {
            int i = tid + t * 256;
            float x = R[(size_t)i * pb + c];
            for (int j = 0; j < c; ++j)
                x -= Q[(size_t)i * n + col0 + j] * dots[j];
            v[t] = x;
            nrm += x * x;
        }
        red[tid] = nrm; __syncthreads();
        for (int s = 128; s > 0; s >>= 1) {
            if (tid < s) red[tid] += red[tid + s];
            __syncthreads();
        }
        float inv = 1.0f / sqrtf(red[0]);
        __syncthreads();
        for (int t = 0; t < per; ++t) {
            int i = tid + t * 256;
            Q[(size_t)i * n + col0 + c] = v[t] * inv;
        }
        __syncthreads();
    }
}

extern "C" void kernel_launch(void* const* d_in, const int* in_sizes, int n_in,
                              void* d_out, int out_size, void* d_ws, size_t ws_size,
                              hipStream_t stream) {
    const float* init_map = (const float*)d_in[0];
    const float* op       = (const float*)d_in[1];

    int n = (int)llround(sqrt((double)in_sizes[0]));   // 2048

    float* T1   = (float*)d_out;                  // temp, overwritten at the end
    float* curr = (float*)d_ws;
    float* Q    = curr + (size_t)n * n;
    float* S    = Q    + (size_t)n * n;
    float* R    = S    + (size_t)n * PB;

    dim3 blk(256);
    dim3 gridBig((n + BN - 1) / BN, (n + BM - 1) / BM);

    // T1 = op @ init_map
    gemm_wmma_f32<<<gridBig, blk, 0, stream>>>(T1, n, op, n, 0, init_map, n,
                                               nullptr, 0, n, n, n, 0);
    // curr = init_map^T @ T1
    gemm_wmma_f32<<<gridBig, blk, 0, stream>>>(curr, n, init_map, n, 1, T1, n,
                                               nullptr, 0, n, n, n, 0);

    // Blocked classical Gram-Schmidt over columns of curr
    for (int p = 0; p < n / PB; ++p) {
        const int kp = p * PB;     // # finished columns == panel start
        if (kp > 0) {
            // S[kp x PB] = Q[:, :kp]^T @ curr[:, panel]
            dim3 gS((PB + BN - 1) / BN, (kp + BM - 1) / BM);
            gemm_wmma_f32<<<gS, blk, 0, stream>>>(S, PB, Q, n, 1,
                                                  curr + kp, n,
                                                  nullptr, 0, kp, PB, n, 0);
        }
        // R[n x PB] = curr[:, panel] - Q[:, :kp] @ S   (K=0 -> plain copy)
        dim3 gR((PB + BN - 1) / BN, (n + BM - 1) / BM);
        gemm_wmma_f32<<<gR, blk, 0, stream>>>(R, PB, Q, n, 0, S, PB,
                                              curr + kp, n, n, PB, kp, 1);
        // sequential CGS inside the panel
        panel_gs<<<1, 256, 0, stream>>>(R, curr, Q, kp, PB, n);
    }

    // out = init_map @ Q
    gemm_wmma_f32<<<gridBig, blk, 0, stream>>>((float*)d_out, n, init_map, n, 0,
                                               Q, n, nullptr, 0, n, n, n, 0);
}